// MambaBlock_without_self_attention_30159260352665
// MI455X (gfx1250) — compile-verified
//
#include <hip/hip_runtime.h>
#include <hip/hip_bf16.h>
#include <math.h>

#define B_SZ   2
#define SEQ    2048
#define HID    1024
#define D_IN   2048
#define N_ST   16
#define DT_R   64
#define KCONV  4
#define EPS    1e-5f
#define NCHUNK 16
#define CHUNK  (SEQ / NCHUNK)   // 128

typedef __attribute__((ext_vector_type(2))) float v2f;
typedef __attribute__((ext_vector_type(8))) float v8f;

__device__ __forceinline__ v8f wmma_f32_16x16x4(v2f a, v2f b, v8f c) {
  // 8 args: (neg_a, A, neg_b, B, c_mod, C, reuse_a, reuse_b)
  return __builtin_amdgcn_wmma_f32_16x16x4_f32(false, a, false, b, (short)0, c,
                                               false, false);
}

__device__ __forceinline__ float gemm_epilogue(float v, int mode,
                                               const float* __restrict__ bias,
                                               int col) {
  if (mode == 1) {                                   // dt_proj: +bias, softplus
    v += bias[col];
    v = (v > 20.f) ? v : __logf(1.f + __expf(v));
  }
  return v;
}

// C[M,N] = A[M,K] @ W[N,K]^T, fp32 WMMA 16x16x4.
// One wave computes a 32x32 tile: 2 A-frags x 2 B-frags -> 4 WMMAs / 4 loads.
__global__ __launch_bounds__(256)
void wmma_gemm_nt(const float* __restrict__ A, int lda,
                  const float* __restrict__ W, int ldw,
                  float* __restrict__ C, int ldc,
                  int M, int ntiles32, int K,
                  int mode, const float* __restrict__ bias)
{
  const int wave = (int)((blockIdx.x * blockDim.x + threadIdx.x) >> 5);
  const int lane = (int)(threadIdx.x & 31);
  const int totalTiles = (M >> 5) * ntiles32;
  if (wave >= totalTiles) return;           // wave-uniform exit
  const int tm   = wave / ntiles32;
  const int tn   = wave % ntiles32;
  const int half = lane >> 4;               // K pair select (0->K0,1 ; 1->K2,3)
  const int mr   = lane & 15;               // M row (A) / N col (B)

  const float* arow0 = A + (size_t)(tm * 32 + mr) * (size_t)lda + 2 * half;
  const float* arow1 = arow0 + (size_t)16 * (size_t)lda;
  const float* brow0 = W + (size_t)(tn * 32 + mr) * (size_t)ldw + 2 * half;
  const float* brow1 = brow0 + (size_t)16 * (size_t)ldw;

  v8f c00 = {}; v8f c01 = {}; v8f c10 = {}; v8f c11 = {};
#pragma unroll 2
  for (int k = 0; k < K; k += 4) {
    v2f a0 = *(const v2f*)(arow0 + k);
    v2f a1 = *(const v2f*)(arow1 + k);
    v2f b0 = *(const v2f*)(brow0 + k);
    v2f b1 = *(const v2f*)(brow1 + k);
    c00 = wmma_f32_16x16x4(a0, b0, c00);
    c01 = wmma_f32_16x16x4(a0, b1, c01);
    c10 = wmma_f32_16x16x4(a1, b0, c10);
    c11 = wmma_f32_16x16x4(a1, b1, c11);
  }

  const int col0 = tn * 32 + mr;
#pragma unroll
  for (int r = 0; r < 8; ++r) {
    const int row0 = tm * 32 + r + 8 * half;
    const int row1 = row0 + 16;
    C[(size_t)row0 * ldc + col0]      = gemm_epilogue(c00[r], mode, bias, col0);
    C[(size_t)row0 * ldc + col0 + 16] = gemm_epilogue(c01[r], mode, bias, col0 + 16);
    C[(size_t)row1 * ldc + col0]      = gemm_epilogue(c10[r], mode, bias, col0);
    C[(size_t)row1 * ldc + col0 + 16] = gemm_epilogue(c11[r], mode, bias, col0 + 16);
  }
}

// Causal depthwise conv (K=4) over time + bias + SiLU.
__global__ __launch_bounds__(256)
void conv_silu_kernel(const float* __restrict__ proj,
                      const float* __restrict__ conv_w,
                      const float* __restrict__ conv_b,
                      float* __restrict__ xt)
{
  const int idx = (int)(blockIdx.x * blockDim.x + threadIdx.x); // (b*SEQ+l)*D_IN+d
  const int d = idx & (D_IN - 1);
  const int l = (idx >> 11) & (SEQ - 1);
  const int b = idx >> 22;
  float acc = conv_b[d];
#pragma unroll
  for (int k = 0; k < KCONV; ++k) {
    const int ls = l - (KCONV - 1) + k;
    if (ls >= 0) {
      const float xv = proj[(size_t)(b * SEQ + ls) * (size_t)(2 * D_IN) + d];
      acc = fmaf(xv, conv_w[d * KCONV + k], acc);
    }
  }
  acc = acc / (1.f + __expf(-acc));        // SiLU
  xt[(size_t)idx] = acc;
}

// ---- Chunk-parallel selective scan (3 phases) ----------------------------
// h_t[n] = exp(dt_t*A[n]) * h_{t-1}[n] + dt_t*x_t*B_t[n]
// Chunk composition factor is exp(A[n] * sum_t dt_t) since A is constant.

// Phase 1: per-chunk local scan from h=0; emit chunk-end state S and sum(dt).
__global__ __launch_bounds__(256)
void scan_pass1(const float* __restrict__ ssm, const float* __restrict__ dtb,
                const float* __restrict__ xt, const float* __restrict__ A_log,
                float* __restrict__ Sbuf, float* __restrict__ sumdt_buf)
{
  const int d = (int)(blockIdx.x * blockDim.x + threadIdx.x);
  const int b = (int)blockIdx.y;
  const int c = (int)blockIdx.z;
  float Av[N_ST], h[N_ST];
#pragma unroll
  for (int n = 0; n < N_ST; ++n) {
    Av[n] = -__expf(A_log[d * N_ST + n]);
    h[n] = 0.f;
  }
  float sumdt = 0.f;
  __shared__ float sB[N_ST];
  for (int t = 0; t < CHUNK; ++t) {
    const size_t row = (size_t)(b * SEQ + c * CHUNK + t);
    if (threadIdx.x < N_ST)
      sB[threadIdx.x] = ssm[row * 96 + DT_R + threadIdx.x];
    __syncthreads();
    const float dt  = dtb[row * D_IN + d];
    const float xs  = xt[row * D_IN + d];
    const float dtx = dt * xs;
    sumdt += dt;
#pragma unroll
    for (int n = 0; n < N_ST; ++n)
      h[n] = __expf(dt * Av[n]) * h[n] + dtx * sB[n];
    __syncthreads();
  }
  const size_t base = ((size_t)(b * NCHUNK + c)) * D_IN + d;
  sumdt_buf[base] = sumdt;
#pragma unroll
  for (int n = 0; n < N_ST; ++n) Sbuf[base * N_ST + n] = h[n];
}

// Phase 2: tiny sequential fix across NCHUNK chunks per (b,d,n) scalar lane.
__global__ __launch_bounds__(256)
void scan_fix(const float* __restrict__ Sbuf, const float* __restrict__ sumdt_buf,
              const float* __restrict__ A_log, float* __restrict__ Hstart)
{
  const int idx = (int)(blockIdx.x * blockDim.x + threadIdx.x); // B*D_IN*N_ST
  const int n = idx & (N_ST - 1);
  const int d = (idx >> 4) & (D_IN - 1);
  const int b = idx >> 15;
  const float Aval = -__expf(A_log[d * N_ST + n]);
  float h = 0.f;
  for (int c = 0; c < NCHUNK; ++c) {
    const size_t base = ((size_t)(b * NCHUNK + c)) * D_IN + d;
    Hstart[base * N_ST + n] = h;
    h = __expf(Aval * sumdt_buf[base]) * h + Sbuf[base * N_ST + n];
  }
}

// Phase 3: re-run chunks in parallel from true Hstart; emit y fused with
// D-skip and SiLU(z) gate.
__global__ __launch_bounds__(256)
void scan_pass2(const float* __restrict__ ssm, const float* __restrict__ dtb,
                const float* __restrict__ xt, const float* __restrict__ proj,
                const float* __restrict__ A_log, const float* __restrict__ Dvec,
                const float* __restrict__ Hstart, float* __restrict__ yb)
{
  const int d = (int)(blockIdx.x * blockDim.x + threadIdx.x);
  const int b = (int)blockIdx.y;
  const int c = (int)blockIdx.z;
  const size_t base = ((size_t)(b * NCHUNK + c)) * D_IN + d;
  float Av[N_ST], h[N_ST];
#pragma unroll
  for (int n = 0; n < N_ST; ++n) {
    Av[n] = -__expf(A_log[d * N_ST + n]);
    h[n] = Hstart[base * N_ST + n];
  }
  const float Dd = Dvec[d];
  __shared__ float sBC[2 * N_ST];
  for (int t = 0; t < CHUNK; ++t) {
    const size_t row = (size_t)(b * SEQ + c * CHUNK + t);
    if (threadIdx.x < 2 * N_ST)
      sBC[threadIdx.x] = ssm[row * 96 + DT_R + threadIdx.x];
    __syncthreads();
    const float dt  = dtb[row * D_IN + d];
    const float xs  = xt[row * D_IN + d];
    const float dtx = dt * xs;
    float y = 0.f;
#pragma unroll
    for (int n = 0; n < N_ST; ++n) {
      h[n] = __expf(dt * Av[n]) * h[n] + dtx * sBC[n];
      y = fmaf(h[n], sBC[N_ST + n], y);
    }
    y = fmaf(Dd, xs, y);
    const float zv = proj[row * (size_t)(2 * D_IN) + D_IN + d];
    y *= zv / (1.f + __expf(-zv));         // * SiLU(z)
    yb[row * D_IN + d] = y;
    __syncthreads();
  }
}

// In-place RMSNorm over HID=1024, one block per row.
__global__ __launch_bounds__(256)
void rmsnorm_kernel(float* __restrict__ out, const float* __restrict__ norm_w)
{
  const int row = (int)blockIdx.x;
  const int tid = (int)threadIdx.x;
  float v[4];
  float ss = 0.f;
#pragma unroll
  for (int i = 0; i < 4; ++i) {
    v[i] = out[(size_t)row * HID + tid * 4 + i];
    ss = fmaf(v[i], v[i], ss);
  }
  __shared__ float red[256];
  red[tid] = ss;
  __syncthreads();
  for (int s = 128; s > 0; s >>= 1) {
    if (tid < s) red[tid] += red[tid + s];
    __syncthreads();
  }
  const float inv = rsqrtf(red[0] / (float)HID + EPS);
#pragma unroll
  for (int i = 0; i < 4; ++i) {
    const int c = tid * 4 + i;
    out[(size_t)row * HID + c] = v[i] * inv * norm_w[c];
  }
}

extern "C" void kernel_launch(void* const* d_in, const int* in_sizes, int n_in,
                              void* d_out, int out_size, void* d_ws, size_t ws_size,
                              hipStream_t stream) {
  const float* hs         = (const float*)d_in[0];
  const float* in_proj_w  = (const float*)d_in[1];
  const float* conv_w     = (const float*)d_in[2];
  const float* conv_b     = (const float*)d_in[3];
  const float* x_proj_w   = (const float*)d_in[4];
  const float* dt_proj_w  = (const float*)d_in[5];
  const float* dt_proj_b  = (const float*)d_in[6];
  const float* A_log      = (const float*)d_in[7];
  const float* Dvec       = (const float*)d_in[8];
  const float* out_proj_w = (const float*)d_in[9];
  const float* norm_w     = (const float*)d_in[10];
  float* out = (float*)d_out;

  const int BL = B_SZ * SEQ;                       // 4096 rows
  float* ws    = (float*)d_ws;
  float* proj  = ws;                               // [BL, 4096]
  float* xt    = proj + (size_t)BL * (2 * D_IN);   // [BL, 2048]
  float* ssm   = xt   + (size_t)BL * D_IN;         // [BL, 96]
  float* dtb   = ssm  + (size_t)BL * 96;           // [BL, 2048]
  float* yb    = dtb  + (size_t)BL * D_IN;         // [BL, 2048]
  float* Sbuf  = yb   + (size_t)BL * D_IN;         // [B,NCHUNK,D_IN,N_ST]
  float* sumdt = Sbuf + (size_t)B_SZ * NCHUNK * D_IN * N_ST; // [B,NCHUNK,D_IN]
  float* Hst   = sumdt + (size_t)B_SZ * NCHUNK * D_IN;       // [B,NCHUNK,D_IN,N_ST]

  // 1) proj = hs @ in_proj_w.T   (M=4096, N=4096, K=1024)
  {
    const int nt = (2 * D_IN) / 32;                // 128
    const int waves = (BL / 32) * nt;              // 16384
    wmma_gemm_nt<<<dim3((waves * 32 + 255) / 256), 256, 0, stream>>>(
        hs, HID, in_proj_w, HID, proj, 2 * D_IN, BL, nt, HID, 0, nullptr);
  }
  // 2) causal depthwise conv + SiLU -> x_t
  conv_silu_kernel<<<(BL * D_IN) / 256, 256, 0, stream>>>(proj, conv_w, conv_b, xt);
  // 3) ssm = x_t @ x_proj_w.T    (M=4096, N=96, K=2048)
  {
    const int nt = 96 / 32;                        // 3
    const int waves = (BL / 32) * nt;              // 384
    wmma_gemm_nt<<<dim3((waves * 32 + 255) / 256), 256, 0, stream>>>(
        xt, D_IN, x_proj_w, D_IN, ssm, 96, BL, nt, D_IN, 0, nullptr);
  }
  // 4) dt = softplus(ssm[:, :64] @ dt_proj_w.T + dt_proj_b)  (N=2048, K=64)
  {
    const int nt = D_IN / 32;                      // 64
    const int waves = (BL / 32) * nt;              // 8192
    wmma_gemm_nt<<<dim3((waves * 32 + 255) / 256), 256, 0, stream>>>(
        ssm, 96, dt_proj_w, DT_R, dtb, D_IN, BL, nt, DT_R, 1, dt_proj_b);
  }
  // 5) chunk-parallel selective scan -> yb
  scan_pass1<<<dim3(D_IN / 256, B_SZ, NCHUNK), 256, 0, stream>>>(
      ssm, dtb, xt, A_log, Sbuf, sumdt);
  scan_fix<<<(B_SZ * D_IN * N_ST) / 256, 256, 0, stream>>>(Sbuf, sumdt, A_log, Hst);
  scan_pass2<<<dim3(D_IN / 256, B_SZ, NCHUNK), 256, 0, stream>>>(
      ssm, dtb, xt, proj, A_log, Dvec, Hst, yb);
  // 6) out = yb @ out_proj_w.T   (M=4096, N=1024, K=2048) -> d_out
  {
    const int nt = HID / 32;                       // 32
    const int waves = (BL / 32) * nt;              // 4096
    wmma_gemm_nt<<<dim3((waves * 32 + 255) / 256), 256, 0, stream>>>(
        yb, D_IN, out_proj_w, D_IN, out, HID, BL, nt, D_IN, 0, nullptr);
  }
  // 7) in-place RMSNorm * norm_w
  rmsnorm_kernel<<<BL, 256, 0, stream>>>(out, norm_w);
}